// GuitarString_19662360281722
// MI455X (gfx1250) — compile-verified
//
#include <hip/hip_runtime.h>

typedef __attribute__((ext_vector_type(2))) float v2f;
typedef __attribute__((ext_vector_type(8))) float v8f;

#define KCHUNK 256

// ---------------------------------------------------------------------------
// Kernel 0: zero impulse response H (workspace) and output accumulator.
// ---------------------------------------------------------------------------
__global__ void ks_zero(float* __restrict__ H, float* __restrict__ acc, int T) {
  int idx = blockIdx.x * blockDim.x + threadIdx.x;
  if (idx < T) { H[idx] = 0.0f; acc[idx] = 0.0f; }
}

// ---------------------------------------------------------------------------
// Kernel 1 (single block): build combined impulse response H so that
//   acc = H (convolve) excitation.
// With C(z)=g_b(a+(1-a)z)((1-b)+bz), D(z)=g_n(a+(1-a)z), P=C*D:
//   H = 0.5*z^{nDown}*[1 + sum_m z^{2mL}P^m + sum_m z^{(2m+1)L} D P^m]
//     + 0.5*z^{nUp}  *[1 + sum_m z^{2mL}P^m + sum_m z^{(2m+1)L} C P^m]
// Q = P^m kept in LDS (deg 3m <= 600), double buffered.
// ---------------------------------------------------------------------------
__global__ void ks_build_impulse(const int* __restrict__ lenp,
                                 const float* __restrict__ pluckp,
                                 const float* __restrict__ nutp,
                                 const float* __restrict__ brgp,
                                 const float* __restrict__ bp,
                                 const float* __restrict__ ap,
                                 float* __restrict__ H, int T) {
  __shared__ float Qa[1024];
  __shared__ float Qb[1024];

  const int   L     = lenp[0];
  const float pluck = pluckp[0];
  const float gn    = nutp[0];
  const float gb    = brgp[0];
  const float b     = bp[0];
  const float a     = ap[0];

  const int nUp   = (int)rintf((float)L * pluck);  // jnp.round = half-to-even
  const int nDown = L - nUp;
  const int niter = T / L;                          // 400

  const float c0 = gb * a * (1.0f - b);
  const float c1 = gb * (a * b + (1.0f - a) * (1.0f - b));
  const float c2 = gb * (1.0f - a) * b;
  const float d0 = gn * a;
  const float d1 = gn * (1.0f - a);
  const float p0 = c0 * d0;
  const float p1 = c0 * d1 + c1 * d0;
  const float p2 = c1 * d1 + c2 * d0;
  const float p3 = c2 * d1;

  const int tid  = threadIdx.x;
  const int nthr = blockDim.x;

  for (int i = tid; i < 1024; i += nthr) { Qa[i] = 0.0f; Qb[i] = 0.0f; }
  __syncthreads();
  if (tid == 0) {
    Qa[0] = 1.0f;                          // Q = P^0
    atomicAdd(&H[nDown], 0.5f);            // identity terms: l0 + r0
    atomicAdd(&H[nUp],   0.5f);
  }
  __syncthreads();

  float* Q  = Qa;
  float* Qn = Qb;
  int deg = 0;                              // deg(Q) = 3m

  for (int m = 0; ; ++m) {
    // even term k = 2m (m >= 1): 0.5 * z^{2mL} P^m on both branch offsets
    if (m >= 1 && 2 * m <= niter) {
      const int off = 2 * m * L;
      for (int j = tid; j <= deg; j += nthr) {
        const float v = 0.5f * Q[j];
        const int dl = off + nDown + j;
        const int dr = off + nUp + j;
        if (dl < T) atomicAdd(&H[dl], v);
        if (dr < T) atomicAdd(&H[dr], v);
      }
    }
    // odd term k = 2m+1: D*P^m hits the l0 (nDown) path, C*P^m the r0 (nUp) path
    if (2 * m + 1 <= niter) {
      const int off = (2 * m + 1) * L;
      for (int j = tid; j <= deg + 2; j += nthr) {
        const float qj  = (j     <= deg)            ? Q[j]     : 0.0f;
        const float qj1 = (j >= 1 && j - 1 <= deg)  ? Q[j - 1] : 0.0f;
        const float qj2 = (j >= 2 && j - 2 <= deg)  ? Q[j - 2] : 0.0f;
        const float dv = d0 * qj + d1 * qj1;            // deg(DQ) = deg+1
        const float cv = c0 * qj + c1 * qj1 + c2 * qj2; // deg(CQ) = deg+2
        const int dl = off + nDown + j;
        const int dr = off + nUp + j;
        if (j <= deg + 1 && dl < T) atomicAdd(&H[dl], 0.5f * dv);
        if (dr < T)                 atomicAdd(&H[dr], 0.5f * cv);
      }
    }
    if (2 * (m + 1) > niter) break;
    // Q <- Q * P
    const int ndeg = deg + 3;
    __syncthreads();
    for (int i = tid; i <= ndeg; i += nthr) {
      const float q0 = (i     <= deg)           ? Q[i]     : 0.0f;
      const float q1 = (i >= 1 && i - 1 <= deg) ? Q[i - 1] : 0.0f;
      const float q2 = (i >= 2 && i - 2 <= deg) ? Q[i - 2] : 0.0f;
      const float q3 = (i >= 3 && i - 3 <= deg) ? Q[i - 3] : 0.0f;
      Qn[i] = p0 * q0 + p1 * q1 + p2 * q2 + p3 * q3;
    }
    __syncthreads();
    float* t = Q; Q = Qn; Qn = t;
    deg = ndeg;
  }
}

// ---------------------------------------------------------------------------
// Kernel 2: lower-triangular Toeplitz matvec acc[t] = sum_d H[d]*x[t-d]
// via V_WMMA_F32_16X16X4_F32, one wave per block, LDS-staged operands.
//   Output tile: D[i][n] covers t = 16*(I0+n) + i, i,n in [0,16).
//   D += T_K * X_{I0-K}, T_K[i][j] = H[16K+i-j], X[j][n] = x[16*(I0+n-K)+j].
//   16x16 T_K split into four 16x4 A-slices (sub = 0..3) -> 4 chained WMMAs.
// A layout (ISA 7.12.2, 32-bit 16x4): M=lane&15; v0 holds K=0|2, v1 holds K=1|3
// (lane halves). B assumed dual: N=lane&15; v0 rows 0|2, v1 rows 1|3.
//
// Boundary handling is done ONCE at staging time (zero-fill H[d<0 or d>=T] and
// negative excitation blocks), so the K loop is fully uniform: no divergence,
// EXEC all-ones, and both operand pairs are ASCENDING LDS address pairs
// (H staged reversed) -> direct ds_load_2addr into operand VGPRs, no swaps.
//   hs: reversed H window, consecutive-word fetch -> conflict-free (dups bcast).
//   xs: stride-17 row padding, lane stride 17 coprime to 64 banks -> no conflict.
// ---------------------------------------------------------------------------
__global__ void ks_toeplitz_wmma(const float* __restrict__ H,
                                 const float* __restrict__ x,
                                 float* __restrict__ acc, int T) {
  __shared__ float hs[16 * KCHUNK + 32];        // 4128 f32 = 16.5 KB (reversed H)
  __shared__ float xs[(KCHUNK + 16) * 17];      // 4624 f32 = 18.5 KB

  const int I0     = blockIdx.x * 16;           // first output block (16 samples)
  const int kstart = blockIdx.y * KCHUNK;
  const int kall   = I0 + 16;                   // K in [0, I0+16)
  const int kend   = (kall < kstart + KCHUNK) ? kall : (kstart + KCHUNK);
  if (kstart >= kend) return;                   // uniform per block: EXEC stays full

  const int lane = threadIdx.x;                 // 0..31
  const int i    = lane & 15;                   // A row M / B col N
  const int hi   = lane >> 4;
  const int kkb  = 2 * hi;                      // K-slot base within v2f

  // ---- stage H window REVERSED: hs[p] = H[htop - p], zero-filling OOB ----
  const int htop   = 16 * kend + 15;            // highest H index needed
  const int hcount = 16 * (kend - kstart) + 32;
  for (int idx = lane; idx < hcount; idx += 32) {
    const int d = htop - idx;
    hs[idx] = (d >= 0 && d < T) ? H[d] : 0.0f;
  }

  // ---- stage excitation blocks [nblk0, nblk0+nrows) into xs (stride 17) ----
  const int nblk0 = I0 - kend + 1;              // may be negative -> zero rows
  const int nrows = (kend - kstart) + 15;       // <= KCHUNK + 15
  for (int idx = lane; idx < nrows * 16; idx += 32) {
    const int r   = idx >> 4;
    const int j   = idx & 15;
    const int blk = nblk0 + r;                  // max blk = I0+15-kstart <= 5999 < T/16
    xs[r * 17 + j] = (blk >= 0) ? x[16 * blk + j] : 0.0f;
  }
  __syncthreads();

  // ---- uniform hot loop: ascending-pair ds_loads + 4 chained WMMAs per K ----
  v8f c = {};
  const int rbias = I0 - nblk0;                 // x row = rbias + i - K
  for (int K = kstart; K < kend; ++K) {
    // A: av.x = H[16K+i-j0] = hs[hrev+j0]; av.y = H[16K+i-j0-1] = hs[hrev+j0+1]
    const float* hp = hs + (htop - 16 * K - i);
    const float* xp = xs + (rbias + i - K) * 17;
#pragma unroll
    for (int sub = 0; sub < 4; ++sub) {
      const int j0 = 4 * sub + kkb;
      v2f av, bv;
      av.x = hp[j0];
      av.y = hp[j0 + 1];
      bv.x = xp[j0];
      bv.y = xp[j0 + 1];
      c = __builtin_amdgcn_wmma_f32_16x16x4_f32(false, av, false, bv,
                                                (short)0, c, false, false);
    }
  }

#pragma unroll
  for (int e = 0; e < 8; ++e) {
    const int t = 16 * (I0 + i) + (e + 8 * hi); // D layout: row = e + 8*hi, col = i
    atomicAdd(&acc[t], c[e]);
  }
}

// ---------------------------------------------------------------------------
extern "C" void kernel_launch(void* const* d_in, const int* in_sizes, int n_in,
                              void* d_out, int out_size, void* d_ws, size_t ws_size,
                              hipStream_t stream) {
  const int*   lenp  = (const int*)d_in[0];
  const float* pluck = (const float*)d_in[1];
  const float* exc   = (const float*)d_in[2];
  const float* nut   = (const float*)d_in[3];
  const float* brg   = (const float*)d_in[4];
  const float* bb    = (const float*)d_in[5];
  const float* da    = (const float*)d_in[6];
  float* acc = (float*)d_out;
  float* H   = (float*)d_ws;                    // T floats of scratch
  const int T = in_sizes[2];                    // 96000

  ks_zero<<<(T + 255) / 256, 256, 0, stream>>>(H, acc, T);
  ks_build_impulse<<<1, 256, 0, stream>>>(lenp, pluck, nut, brg, bb, da, H, T);

  const int ntiles = (T + 255) / 256;                       // 375 output tiles
  const int nkc    = (T / 16 + KCHUNK - 1) / KCHUNK;        // K chunks (24)
  dim3 grid(ntiles, nkc);
  ks_toeplitz_wmma<<<grid, 32, 0, stream>>>(H, exc, acc, T);
}